// GTN_34583076668022
// MI455X (gfx1250) — compile-verified
//
#include <hip/hip_runtime.h>
#include <hip/hip_bf16.h>

// GTN on MI455X (gfx1250). Fully-linear network -> collapse both layers into:
//   X0 = [S0+x | A | deg+1]  (N x 208, padded)   h   = X0 @ Wcat0 + b0
//   X1 = [S1+h | A | deg+1]  (N x 192, padded)   out = X1 @ Wfin  + bfin
// where S*[dst] += feats[src] (edge scatter), A[dst] += edge_attr[e] (once).
// GEMMs use V_WMMA_F32_16X16X4_F32 (full fp32 precision, wave32).

#define GN 50000
#define GE 800000
#define INCH 151
#define HIDD 128
#define OUTD 51
#define EDIM 51
#define K0P 208   // 151 + 51 + 1 -> pad to mult of 16
#define K1P 192   // 128 + 51 + 1 -> pad to mult of 16

typedef __attribute__((ext_vector_type(2))) float v2f;
typedef __attribute__((ext_vector_type(8))) float v8f;

static __device__ __forceinline__ void atomAddF(float* p, float v) {
  __hip_atomic_fetch_add(p, v, __ATOMIC_RELAXED, __HIP_MEMORY_SCOPE_AGENT);
}

// ---------------- weight prep (tiny, one block each) ----------------
// Wpk0: Wcat0 [208 x 128] stored k-pair interleaved: Wpk0[((r>>1)*128+c)*2+(r&1)]
//   rows 0..150  : W0
//   rows 151..201: We0 @ W0
//   row  202     : be0 @ W0
// W1W: W1 @ Wout [128 x 51];  bfin[64] = b1 @ Wout + bout (cols >=51 zero)
__global__ void prep1(const float* __restrict__ We0, const float* __restrict__ W0,
                      const float* __restrict__ be0, const float* __restrict__ W1,
                      const float* __restrict__ Wout, const float* __restrict__ b1,
                      const float* __restrict__ bout,
                      float* __restrict__ Wpk0, float* __restrict__ W1W,
                      float* __restrict__ bfin) {
  for (int idx = threadIdx.x; idx < K0P * HIDD; idx += 256) {
    int r = idx >> 7, c = idx & 127;
    float v = 0.f;
    if (r < INCH) {
      v = W0[r * HIDD + c];
    } else if (r < INCH + EDIM) {
      int rr = r - INCH;
      for (int k = 0; k < INCH; ++k) v += We0[rr * INCH + k] * W0[k * HIDD + c];
    } else if (r == INCH + EDIM) {
      for (int k = 0; k < INCH; ++k) v += be0[k] * W0[k * HIDD + c];
    }
    Wpk0[((r >> 1) * HIDD + c) * 2 + (r & 1)] = v;
  }
  for (int idx = threadIdx.x; idx < HIDD * OUTD; idx += 256) {
    int r = idx / OUTD, c = idx - r * OUTD;
    float v = 0.f;
    for (int k = 0; k < HIDD; ++k) v += W1[r * HIDD + k] * Wout[k * OUTD + c];
    W1W[idx] = v;
  }
  if (threadIdx.x < 64) {
    int c = threadIdx.x;
    float v = 0.f;
    if (c < OUTD) {
      for (int k = 0; k < HIDD; ++k) v += b1[k] * Wout[k * OUTD + c];
      v += bout[c];
    }
    bfin[c] = v;
  }
}

// Wfin: [192 x 64] pair-interleaved; rows 0..127: W1W; 128..178: We1@W1W;
// row 179: be1@W1W; rest (and cols>=51) zero.
__global__ void prep2(const float* __restrict__ We1, const float* __restrict__ be1,
                      const float* __restrict__ W1W, float* __restrict__ Wfin) {
  for (int idx = threadIdx.x; idx < K1P * 64; idx += 256) {
    int r = idx >> 6, c = idx & 63;
    float v = 0.f;
    if (c < OUTD) {
      if (r < HIDD) {
        v = W1W[r * OUTD + c];
      } else if (r < HIDD + EDIM) {
        int rr = r - HIDD;
        for (int k = 0; k < HIDD; ++k) v += We1[rr * HIDD + k] * W1W[k * OUTD + c];
      } else if (r == HIDD + EDIM) {
        for (int k = 0; k < HIDD; ++k) v += be1[k] * W1W[k * OUTD + c];
      }
    }
    Wfin[((r >> 1) * 64 + c) * 2 + (r & 1)] = v;
  }
}

// ---------------- edge scatter passes ----------------
// One wave (32 lanes) per edge; columns strided across lanes.
__global__ void edge_scatter1(const float* __restrict__ x, const float* __restrict__ ea,
                              const int* __restrict__ ei, float* __restrict__ X0) {
  int e = (blockIdx.x * blockDim.x + threadIdx.x) >> 5;
  int lane = threadIdx.x & 31;
  if (e >= GE) return;
  int src = ei[e];
  int dst = ei[GE + e];
  const float* xr = x + (size_t)src * INCH;
  float* o = X0 + (size_t)dst * K0P;
  for (int c = lane; c < INCH; c += 32) atomAddF(&o[c], xr[c]);
  const float* er = ea + (size_t)e * EDIM;
  for (int f = lane; f < EDIM; f += 32) atomAddF(&o[INCH + f], er[f]);
  if (lane == 0) atomAddF(&o[INCH + EDIM], 1.0f);  // deg
}

__global__ void edge_scatter2(const float* __restrict__ h, const int* __restrict__ ei,
                              float* __restrict__ X1) {
  int e = (blockIdx.x * blockDim.x + threadIdx.x) >> 5;
  int lane = threadIdx.x & 31;
  if (e >= GE) return;
  int src = ei[e];
  int dst = ei[GE + e];
  const float* hr = h + (size_t)src * HIDD;
  float* o = X1 + (size_t)dst * K1P;
  #pragma unroll
  for (int c = lane; c < HIDD; c += 32) atomAddF(&o[c], hr[c]);
}

// ---------------- node self-terms ----------------
__global__ void node_self0(const float* __restrict__ x, float* __restrict__ X0) {
  int t = blockIdx.x * 256 + threadIdx.x;            // exactly GN*K0P threads
  int n = t / K0P, c = t - n * K0P;
  if (c < INCH) X0[t] += x[(size_t)n * INCH + c];
  else if (c == INCH + EDIM) X0[t] += 1.0f;          // deg -> deg+1
}

__global__ void node_self1(const float* __restrict__ h, const float* __restrict__ X0,
                           float* __restrict__ X1) {
  int t = blockIdx.x * 256 + threadIdx.x;            // exactly GN*K1P threads
  int n = t / K1P, c = t - n * K1P;
  if (c < HIDD) X1[t] += h[(size_t)n * HIDD + c];
  else if (c < HIDD + EDIM + 1)                      // copy [A | deg+1] from X0
    X1[t] = X0[(size_t)n * K0P + (INCH - HIDD) + c]; // 151 + (c-128) = c + 23
}

// ---------------- WMMA fp32 GEMM: out[N x outCols] = X[N x K] @ Wpk + bias ----
// One wave computes a 16x64 output strip via 4 accumulators of 16x16.
// A-lane layout (16x4 f32): m=lane&15, k0=k+2*(lane>>4), holds {X[m][k0],X[m][k0+1]}.
// B pre-packed so lane loads {W[k0][col],W[k0+1][col]} as one float2.
__global__ void gemm_wmma(const float* __restrict__ X, const float* __restrict__ Wpk,
                          const float* __restrict__ bias, float* __restrict__ out,
                          int rowTiles, int colGroups, int K, int NC,
                          int ldOut, int outCols) {
  int waveId = threadIdx.x >> 5;
  int lane = threadIdx.x & 31;
  int tile = blockIdx.x * (blockDim.x >> 5) + waveId;
  if (tile >= rowTiles * colGroups) return;          // wave-uniform
  int rowTile = tile / colGroups;
  int colGroup = tile - rowTile * colGroups;

  int m = lane & 15;
  int kph = lane >> 4;                               // 0 or 1 (upper K pair)
  const float* xrow = X + (size_t)(rowTile * 16 + m) * K;
  const v2f* wp = (const v2f*)Wpk;
  int cb = colGroup * 64 + m;

  v8f acc[4] = {v8f{}, v8f{}, v8f{}, v8f{}};
  for (int k = 0; k < K; k += 4) {
    int k0 = k + (kph << 1);
    v2f a = *(const v2f*)(xrow + k0);
    const v2f* wb = wp + (size_t)(k0 >> 1) * NC + cb;
    #pragma unroll
    for (int j = 0; j < 4; ++j) {
      v2f b = wb[j * 16];
      // D = A(16x4) x B(4x16) + C, fp32 throughout
      acc[j] = __builtin_amdgcn_wmma_f32_16x16x4_f32(
          false, a, false, b, (short)0, acc[j], false, false);
    }
  }

  int rbase = rowTile * 16 + (kph << 3);             // lanes 16-31 -> rows M=8..15
  #pragma unroll
  for (int j = 0; j < 4; ++j) {
    int col = colGroup * 64 + j * 16 + m;
    if (col < outCols) {
      float bv = bias ? bias[col] : 0.0f;
      float* op = out + (size_t)rbase * ldOut + col;
      #pragma unroll
      for (int r = 0; r < 8; ++r) op[(size_t)r * ldOut] = acc[j][r] + bv;
    }
  }
}

extern "C" void kernel_launch(void* const* d_in, const int* in_sizes, int n_in,
                              void* d_out, int out_size, void* d_ws, size_t ws_size,
                              hipStream_t stream) {
  const float* x    = (const float*)d_in[0];
  const float* ea   = (const float*)d_in[1];
  const float* We0  = (const float*)d_in[2];
  const float* be0  = (const float*)d_in[3];
  const float* W0   = (const float*)d_in[4];
  const float* b0   = (const float*)d_in[5];
  const float* We1  = (const float*)d_in[6];
  const float* be1  = (const float*)d_in[7];
  const float* W1   = (const float*)d_in[8];
  const float* b1   = (const float*)d_in[9];
  const float* Wout = (const float*)d_in[10];
  const float* bout = (const float*)d_in[11];
  const int*   ei   = (const int*)d_in[12];

  float* ws   = (float*)d_ws;
  float* X0   = ws;                               // N*208
  float* X1   = X0 + (size_t)GN * K0P;            // N*192
  float* h    = X1 + (size_t)GN * K1P;            // N*128
  float* Wpk0 = h + (size_t)GN * HIDD;            // 208*128
  float* Wfin = Wpk0 + K0P * HIDD;                // 192*64
  float* W1W  = Wfin + K1P * 64;                  // 128*51
  float* bfin = W1W + HIDD * OUTD;                // 64

  hipMemsetAsync(X0, 0, (size_t)GN * K0P * sizeof(float), stream);
  hipMemsetAsync(X1, 0, (size_t)GN * K1P * sizeof(float), stream);

  prep1<<<1, 256, 0, stream>>>(We0, W0, be0, W1, Wout, b1, bout, Wpk0, W1W, bfin);
  prep2<<<1, 256, 0, stream>>>(We1, be1, W1W, Wfin);

  edge_scatter1<<<GE * 32 / 256, 256, 0, stream>>>(x, ea, ei, X0);
  node_self0<<<GN * K0P / 256, 256, 0, stream>>>(x, X0);

  // h = X0 @ Wcat0 + b0 : rowTiles=3125, colGroups=2 (128 cols)
  {
    int tiles = 3125 * 2;
    gemm_wmma<<<(tiles + 7) / 8, 256, 0, stream>>>(X0, Wpk0, b0, h,
                                                   3125, 2, K0P, HIDD, HIDD, HIDD);
  }

  edge_scatter2<<<GE * 32 / 256, 256, 0, stream>>>(h, ei, X1);
  node_self1<<<GN * K1P / 256, 256, 0, stream>>>(h, X0, X1);

  // out = X1 @ Wfin + bfin : rowTiles=3125, colGroups=1 (64 padded cols, store 51)
  {
    int tiles = 3125;
    gemm_wmma<<<(tiles + 7) / 8, 256, 0, stream>>>(X1, Wfin, bfin, (float*)d_out,
                                                   3125, 1, K1P, 64, OUTD, OUTD);
  }
}